// DPhysics_63462436766077
// MI455X (gfx1250) — compile-verified
//
#include <hip/hip_runtime.h>

// ---------------------------------------------------------------------------
// Problem constants (match reference)
// ---------------------------------------------------------------------------
namespace {
constexpr int   NB = 64;     // batches
constexpr int   NT = 500;    // timesteps
constexpr int   NP = 128;    // robot points
constexpr int   GH = 256;    // grid H
constexpr int   GW = 256;    // grid W
constexpr float MASS_C      = 40.0f;
constexpr float GRAV_C      = 9.8f;
constexpr float STIFF_C     = 1000.0f;
constexpr float DAMP_C      = 100.0f;
constexpr float OMEGA_MAX_C = 2.0f;
constexpr float DT_C        = 0.01f;
constexpr float LYH_C       = 0.3f;        // LY/2
constexpr float INV_RES_C   = 10.0f;       // 1/RES
constexpr float HALF_SPAN_C = 12.8f;       // GH*RES/2
constexpr float FCLAMP_C    = MASS_C * GRAV_C;  // 392
}

typedef __attribute__((ext_vector_type(2))) float v2f;
typedef __attribute__((ext_vector_type(8))) float v8f;

__device__ __forceinline__ float wave_sum32(float v) {
#pragma unroll
  for (int m = 16; m >= 1; m >>= 1) v += __shfl_xor(v, m, 32);
  return v;
}
__device__ __forceinline__ float clampf(float v, float c) {
  return fminf(fmaxf(v, -c), c);
}

// ---------------------------------------------------------------------------
// Kernel 1: inertia-inverse precompute via V_WMMA_F32_16X16X4_F32.
// Moment matrix M(j,k) = sum_n mp * P[n][j] * P[n][k]  (3x128 @ 128x3),
// done as 32 accumulating 16x16x4 f32 WMMAs (K chunks of 4 points).
// A(16x4): A[m][k] = mp*P[pt][m] (rows >=3 zero);  B(4x16): B[k][n] = P[pt][n].
// Per ISA layout: lane half selects K block {0,1}/{2,3}; m=n=lane&15.
// Then I = tr(M)*I3 - M, inverted analytically; 9 floats -> d_ws.
// One wave of 32 lanes => EXEC all ones at the WMMA (required).
// ---------------------------------------------------------------------------
__global__ void DPhys_inertia_wmma(const float* __restrict__ P,
                                   float* __restrict__ iinv) {
  const int lane = threadIdx.x;        // 0..31
  const int m    = lane & 15;          // row of A / col of B
  const int kb   = (lane >> 4) * 2;    // K block base: 0 or 2
  const float mp = MASS_C / (float)NP;

  v8f acc = {};
#pragma unroll 4
  for (int c = 0; c < NP / 4; ++c) {
    const int p0 = c * 4 + kb;         // point index for K=kb
    v2f a = {};
    v2f b = {};
    if (m < 3) {
      a.x = mp * P[p0 * 3 + m];
      a.y = mp * P[(p0 + 1) * 3 + m];
      b.x = P[p0 * 3 + m];
      b.y = P[(p0 + 1) * 3 + m];
    }
    acc = __builtin_amdgcn_wmma_f32_16x16x4_f32(
        /*neg_a=*/false, a, /*neg_b=*/false, b,
        /*c_mod=*/(short)0, acc, /*reuse_a=*/false, /*reuse_b=*/false);
  }

  // C/D layout: VGPR r -> (M=r, N=lane) for lanes 0..15. We need M(0..2, 0..2):
  __shared__ float Msh[3][3];
  if (lane < 3) {
    Msh[0][lane] = acc[0];
    Msh[1][lane] = acc[1];
    Msh[2][lane] = acc[2];
  }
  __syncthreads();
  if (lane == 0) {
    const float tr  = Msh[0][0] + Msh[1][1] + Msh[2][2];
    const float a   = tr - Msh[0][0];   // Ixx
    const float d   = tr - Msh[1][1];   // Iyy
    const float f   = tr - Msh[2][2];   // Izz
    const float b   = -Msh[0][1];       // Ixy
    const float c   = -Msh[0][2];       // Ixz
    const float e   = -Msh[1][2];       // Iyz
    const float c00 = d * f - e * e;
    const float c01 = c * e - b * f;
    const float c02 = b * e - c * d;
    const float det = a * c00 + b * c01 + c * c02;
    const float di  = 1.0f / det;
    iinv[0] = c00 * di;  iinv[1] = c01 * di;  iinv[2] = c02 * di;
    iinv[3] = c01 * di;  iinv[4] = (a * f - c * c) * di;  iinv[5] = (b * c - a * e) * di;
    iinv[6] = c02 * di;  iinv[7] = (b * c - a * e) * di;  iinv[8] = (a * d - b * b) * di;
  }
}

// ---------------------------------------------------------------------------
// Kernel 2: the rollout. One block per batch, 128 threads = 1 point each.
// Per-batch scalar state replicated in registers of every thread; reductions
// over the 128 points via wave32 shuffle butterflies + 4-slot LDS combine.
// Grids (32 MB total) stay L2-resident; all gathers hit L2 after warmup.
// ---------------------------------------------------------------------------
__global__ __launch_bounds__(128) void DPhys_rollout(
    const float* __restrict__ z_grid, const float* __restrict__ friction,
    const float* __restrict__ controls, const float* __restrict__ Pts,
    const float* __restrict__ x0g, const float* __restrict__ xd0g,
    const float* __restrict__ R0g, const float* __restrict__ om0g,
    const float* __restrict__ iinv_g, float* __restrict__ out) {
  const int bb   = blockIdx.x;
  const int tid  = threadIdx.x;    // point index 0..127
  const int lane = tid & 31;
  const int wid  = tid >> 5;

  __shared__ float redA[4];        // in-contact weight partials
  __shared__ float redB[4][8];     // force(3)+torque(3) partials (padded)

  const float* zb = z_grid   + (size_t)bb * (GH * GW);
  const float* fb = friction + (size_t)bb * (GH * GW);

  // per-point constants
  const float ppx = Pts[tid * 3 + 0];
  const float ppy = Pts[tid * 3 + 1];
  const float ppz = Pts[tid * 3 + 2];
  const float mL  = (ppy < 0.0f) ? 1.0f : 0.0f;   // maskL

  // per-batch state (identical in every thread)
  float xx = x0g[bb * 3 + 0], xy = x0g[bb * 3 + 1], xz = x0g[bb * 3 + 2];
  float xdx = xd0g[bb * 3 + 0], xdy = xd0g[bb * 3 + 1], xdz = xd0g[bb * 3 + 2];
  float r00 = R0g[bb * 9 + 0], r01 = R0g[bb * 9 + 1], r02 = R0g[bb * 9 + 2];
  float r10 = R0g[bb * 9 + 3], r11 = R0g[bb * 9 + 4], r12 = R0g[bb * 9 + 5];
  float r20 = R0g[bb * 9 + 6], r21 = R0g[bb * 9 + 7], r22 = R0g[bb * 9 + 8];
  float omx = om0g[bb * 3 + 0], omy = om0g[bb * 3 + 1], omz = om0g[bb * 3 + 2];
  const float J0 = iinv_g[0], J1 = iinv_g[1], J2 = iinv_g[2];
  const float J3 = iinv_g[3], J4 = iinv_g[4], J5 = iinv_g[5];
  const float J6 = iinv_g[6], J7 = iinv_g[7], J8 = iinv_g[8];

#pragma unroll 1
  for (int t = 0; t < NT; ++t) {
    // ---- controls / thrust -------------------------------------------------
    const float cv = controls[((size_t)bb * NT + t) * 2 + 0];
    const float cw = controls[((size_t)bb * NT + t) * 2 + 1];
    const float vLs = cv - cw * LYH_C, vRs = cv + cw * LYH_C;
    float tnx = r00, tny = r10, tnz = r20;                  // R[...,0]
    const float tin = 1.0f / fmaxf(sqrtf(tnx*tnx + tny*tny + tnz*tnz), 1e-6f);
    tnx *= tin; tny *= tin; tnz *= tin;

    // ---- point kinematics --------------------------------------------------
    const float relx = r00 * ppx + r01 * ppy + r02 * ppz;   // R @ p
    const float rely = r10 * ppx + r11 * ppy + r12 * ppz;
    const float relz = r20 * ppx + r21 * ppy + r22 * ppz;
    const float ptx = relx + xx, pty = rely + xy, ptz = relz + xz;
    const float vx = xdx + (omy * relz - omz * rely);
    const float vy = xdy + (omz * relx - omx * relz);
    const float vz = xdz + (omx * rely - omy * relx);

    // ---- bilinear sample of z, gx, gy, friction ---------------------------
    const float uf = fminf(fmaxf((ptx + HALF_SPAN_C) * INV_RES_C, 0.0f), 254.999f);
    const float vf = fminf(fmaxf((pty + HALF_SPAN_C) * INV_RES_C, 0.0f), 254.999f);
    const int u0 = (int)uf, v0 = (int)vf;
    const float du = uf - (float)u0, dv = vf - (float)v0;
    const int um = max(u0 - 1, 0), up = min(u0 + 2, GH - 1);
    const int vm = max(v0 - 1, 0), vp = min(v0 + 2, GW - 1);
    const int i0 = u0 * GW, i1 = (u0 + 1) * GW;
    const float z00 = zb[i0 + v0],     z01 = zb[i0 + v0 + 1];
    const float z10 = zb[i1 + v0],     z11 = zb[i1 + v0 + 1];
    const float zm0 = zb[um * GW + v0], zm1 = zb[um * GW + v0 + 1];
    const float zp0 = zb[up * GW + v0], zp1 = zb[up * GW + v0 + 1];
    const float z0m = zb[i0 + vm],     z1m = zb[i1 + vm];
    const float z0p = zb[i0 + vp],     z1p = zb[i1 + vp];
    const float f00 = fb[i0 + v0],     f01 = fb[i0 + v0 + 1];
    const float f10 = fb[i1 + v0],     f11 = fb[i1 + v0 + 1];
    // jnp.gradient: central interior, one-sided edges, then /RES
    const float sxm = INV_RES_C / (float)(u0 + 1 - um);
    const float sxp = INV_RES_C / (float)(up - u0);
    const float sym = INV_RES_C / (float)(v0 + 1 - vm);
    const float syp = INV_RES_C / (float)(vp - v0);
    const float gx00 = (z10 - zm0) * sxm, gx01 = (z11 - zm1) * sxm;
    const float gx10 = (zp0 - z00) * sxp, gx11 = (zp1 - z01) * sxp;
    const float gy00 = (z01 - z0m) * sym, gy10 = (z11 - z1m) * sym;
    const float gy01 = (z0p - z00) * syp, gy11 = (z1p - z10) * syp;
    const float w00 = (1.0f - du) * (1.0f - dv), w10 = du * (1.0f - dv);
    const float w01 = (1.0f - du) * dv,          w11 = du * dv;
    const float zs  = w00 * z00  + w10 * z10  + w01 * z01  + w11 * z11;
    const float gxs = w00 * gx00 + w10 * gx10 + w01 * gx01 + w11 * gx11;
    const float gys = w00 * gy00 + w10 * gy10 + w01 * gy01 + w11 * gy11;
    const float frs = w00 * f00  + w10 * f10  + w01 * f01  + w11 * f11;

    // ---- contact model -----------------------------------------------------
    float nx = -gxs, ny = -gys, nz = 1.0f;
    const float nin = 1.0f / fmaxf(sqrtf(nx * nx + ny * ny + 1.0f), 1e-6f);
    nx *= nin; ny *= nin; nz *= nin;
    const float dh  = ptz - zs;
    const float ic  = 1.0f / (1.0f + __expf(10.0f * dh));
    const float xdn = vx * nx + vy * ny + vz * nz;
    const float fm  = -(STIFF_C * dh + DAMP_C * xdn);
    const float Fpx = fm * nx, Fpy = fm * ny, Fpz = fm * nz;

    // reduce sum(in_c) over 128 points
    const float icw = wave_sum32(ic);
    if (lane == 0) redA[wid] = icw;
    __syncthreads();
    const float sc = ic / (redA[0] + redA[1] + redA[2] + redA[3]);

    const float Frx = clampf(Fpx * sc, FCLAMP_C);
    const float Fry = clampf(Fpy * sc, FCLAMP_C);
    const float Frz = clampf(Fpz * sc, FCLAMP_C);
    const float Nmag = sqrtf(Frx * Frx + Fry * Fry + Frz * Frz);

    const float spd = mL * vLs + (1.0f - mL) * vRs;
    const float slx = frs * (spd * tnx - vx);
    const float sly = frs * (spd * tny - vy);
    const float slz = frs * (spd * tnz - vz);
    const float sdn = slx * nx + sly * ny + slz * nz;
    const float Ffx = clampf(Nmag * (slx - sdn * nx), FCLAMP_C);
    const float Ffy = clampf(Nmag * (sly - sdn * ny), FCLAMP_C);
    const float Ffz = clampf(Nmag * (slz - sdn * nz), FCLAMP_C);

    float l6[6];
    l6[0] = Frx + Ffx; l6[1] = Fry + Ffy; l6[2] = Frz + Ffz;
    l6[3] = rely * l6[2] - relz * l6[1];   // torque = rel x F
    l6[4] = relz * l6[0] - relx * l6[2];
    l6[5] = relx * l6[1] - rely * l6[0];
#pragma unroll
    for (int i = 0; i < 6; ++i) {
      const float r = wave_sum32(l6[i]);
      if (lane == 0) redB[wid][i] = r;
    }
    __syncthreads();
    float s6[6];
#pragma unroll
    for (int i = 0; i < 6; ++i)
      s6[i] = redB[0][i] + redB[1][i] + redB[2][i] + redB[3][i];

    // ---- integrate (identical on all threads) ------------------------------
    const float odx = clampf(J0 * s6[3] + J1 * s6[4] + J2 * s6[5], OMEGA_MAX_C);
    const float ody = clampf(J3 * s6[3] + J4 * s6[4] + J5 * s6[5], OMEGA_MAX_C);
    const float odz = clampf(J6 * s6[3] + J7 * s6[4] + J8 * s6[5], OMEGA_MAX_C);
    xdx += (s6[0] * (1.0f / MASS_C)) * DT_C;
    xdy += (s6[1] * (1.0f / MASS_C)) * DT_C;
    xdz += ((s6[2] - MASS_C * GRAV_C) * (1.0f / MASS_C)) * DT_C;
    xx += xdx * DT_C; xy += xdy * DT_C; xz += xdz * DT_C;
    omx += odx * DT_C; omy += ody * DT_C; omz += odz * DT_C;

    // Rodrigues: R = R @ (I + sin(th dt) K + (1-cos(th dt)) K^2), K = skew(w)/max(th,eps)
    const float th  = sqrtf(omx * omx + omy * omy + omz * omz);
    const float thi = 1.0f / fmaxf(th, 1e-6f);
    const float k0 = omx * thi, k1 = omy * thi, k2 = omz * thi;
    const float sa = __sinf(th * DT_C);
    const float ca = 1.0f - __cosf(th * DT_C);
    const float q00 = 1.0f - ca * (k1 * k1 + k2 * k2);
    const float q01 = -sa * k2 + ca * k0 * k1;
    const float q02 =  sa * k1 + ca * k0 * k2;
    const float q10 =  sa * k2 + ca * k0 * k1;
    const float q11 = 1.0f - ca * (k0 * k0 + k2 * k2);
    const float q12 = -sa * k0 + ca * k1 * k2;
    const float q20 = -sa * k1 + ca * k0 * k2;
    const float q21 =  sa * k0 + ca * k1 * k2;
    const float q22 = 1.0f - ca * (k0 * k0 + k1 * k1);
    float n00 = r00 * q00 + r01 * q10 + r02 * q20;
    float n01 = r00 * q01 + r01 * q11 + r02 * q21;
    float n02 = r00 * q02 + r01 * q12 + r02 * q22;
    float n10 = r10 * q00 + r11 * q10 + r12 * q20;
    float n11 = r10 * q01 + r11 * q11 + r12 * q21;
    float n12 = r10 * q02 + r11 * q12 + r12 * q22;
    float n20 = r20 * q00 + r21 * q10 + r22 * q20;
    float n21 = r20 * q01 + r21 * q11 + r22 * q21;
    float n22 = r20 * q02 + r21 * q12 + r22 * q22;
    r00 = n00; r01 = n01; r02 = n02;
    r10 = n10; r11 = n11; r12 = n12;
    r20 = n20; r21 = n21; r22 = n22;

    if (tid == 0) {
      float* o = out + ((size_t)bb * NT + t) * 3;
      o[0] = xx; o[1] = xy; o[2] = xz;
    }
  }
}

// ---------------------------------------------------------------------------
// Entry point
// ---------------------------------------------------------------------------
extern "C" void kernel_launch(void* const* d_in, const int* in_sizes, int n_in,
                              void* d_out, int out_size, void* d_ws, size_t ws_size,
                              hipStream_t stream) {
  (void)in_sizes; (void)n_in; (void)out_size; (void)ws_size;
  const float* z_grid       = (const float*)d_in[0];
  const float* friction     = (const float*)d_in[1];
  const float* controls     = (const float*)d_in[2];
  const float* robot_points = (const float*)d_in[3];
  const float* x0           = (const float*)d_in[4];
  const float* xd0          = (const float*)d_in[5];
  const float* R0           = (const float*)d_in[6];
  const float* omega0       = (const float*)d_in[7];
  float* iinv = (float*)d_ws;   // 9 floats of scratch

  DPhys_inertia_wmma<<<1, 32, 0, stream>>>(robot_points, iinv);
  DPhys_rollout<<<NB, 128, 0, stream>>>(z_grid, friction, controls, robot_points,
                                        x0, xd0, R0, omega0, iinv,
                                        (float*)d_out);
}